// S2_to_S1_67568425501191
// MI455X (gfx1250) — compile-verified
//
#include <hip/hip_runtime.h>
#include <hip/hip_bf16.h>

// ---------------------------------------------------------------------------
// Shapes (fixed by the reference)
// ---------------------------------------------------------------------------
#define NB   128   // N
#define DD   128   // D
#define TT   256   // T
#define VV   7     // V
#define WW   3     // W
#define DIMK 128   // DIM

typedef float v2f __attribute__((ext_vector_type(2)));
typedef float v8f __attribute__((ext_vector_type(8)));

// ---------------------------------------------------------------------------
// K1: mean over T.  x:(N,D,T,nn) -> xm:(N*nn, D) row-major
// ---------------------------------------------------------------------------
__global__ void mean_over_t(const float* __restrict__ x, float* __restrict__ xm,
                            int nn, int total) {
    int idx = blockIdx.x * blockDim.x + threadIdx.x;
    if (idx >= total) return;                  // total = N*D*nn
    int n = idx / (DD * nn);
    int r = idx % (DD * nn);
    int d = r / nn;
    int v = r % nn;
    const float* p = x + ((size_t)(n * DD + d) * TT) * nn + v;
    float s = 0.f;
    #pragma unroll 8
    for (int t = 0; t < TT; ++t) s += p[(size_t)t * nn];
    xm[(size_t)(n * nn + v) * DD + d] = s * (1.0f / (float)TT);
}

// ---------------------------------------------------------------------------
// K2: f32 WMMA GEMM: C[M,N] = A[M,K] @ B[K,N], optional ReLU.
// One wave per 16x16 tile.  Uses V_WMMA_F32_16X16X4_F32 (wave32, EXEC all 1s).
// A 16x4 layout: lane -> M=lane&15, K pair selected by lane>=16 (K= half*2 +vgpr)
// B 4x16 layout (mirror): lane -> N=lane&15, K = half*2 + vgpr
// C/D 16x16: VGPR g -> row = g + half*8, col = lane&15
// ---------------------------------------------------------------------------
__global__ __launch_bounds__(32) void gemm_wmma_f32(
        const float* __restrict__ A, const float* __restrict__ B,
        float* __restrict__ C, int M, int K, int N, int do_relu) {
    const int mt   = blockIdx.x;
    const int nt   = blockIdx.y;
    const int lane = threadIdx.x;
    const int half = lane >> 4;
    const int l16  = lane & 15;

    const float* __restrict__ Arow = A + (size_t)(mt * 16 + l16) * K;
    const float* __restrict__ Bcol = B + nt * 16 + l16;

    v8f acc = {0.f, 0.f, 0.f, 0.f, 0.f, 0.f, 0.f, 0.f};
    for (int k0 = 0; k0 < K; k0 += 4) {
        v2f a, b;
        const int ka = k0 + half * 2;
        a.x = Arow[ka + 0];
        a.y = Arow[ka + 1];
        b.x = Bcol[(size_t)(ka + 0) * N];
        b.y = Bcol[(size_t)(ka + 1) * N];
        acc = __builtin_amdgcn_wmma_f32_16x16x4_f32(
                  false, a, false, b, (short)0, acc, false, false);
    }
    float* __restrict__ Crow = C + nt * 16 + l16;
    #pragma unroll
    for (int g = 0; g < 8; ++g) {
        float v = acc[g];
        if (do_relu) v = fmaxf(v, 0.f);
        Crow[(size_t)(mt * 16 + g + half * 8) * N] = v;
    }
}

// ---------------------------------------------------------------------------
// K3: edge input: ei[n,e,k] = h[n,i_e,k] * h[n,j_e,k]
// edge ordering of _rel_onehot: e-th edge has i = e/(nn-1), j skips i.
// ---------------------------------------------------------------------------
__global__ void edge_build(const float* __restrict__ h, float* __restrict__ ei,
                           int nn, int total) {
    int idx = blockIdx.x * blockDim.x + threadIdx.x;
    if (idx >= total) return;                  // total = N*E*DIM
    int E = nn * (nn - 1);
    int n = idx / (E * DIMK);
    int e = (idx / DIMK) % E;
    int k = idx % DIMK;
    int i = e / (nn - 1);
    int r = e % (nn - 1);
    int j = (r < i) ? r : r + 1;
    ei[idx] = h[(size_t)(n * nn + i) * DIMK + k] *
              h[(size_t)(n * nn + j) * DIMK + k];
}

// ---------------------------------------------------------------------------
// K4: receiver aggregation: hsum[n,v,k] = h[n,v,k] + sum_{e: recv=v} edge[n,e,k]
// (edges with receiver v are contiguous: e in [v*(nn-1), (v+1)*(nn-1)))
// ---------------------------------------------------------------------------
__global__ void agg_edges(const float* __restrict__ h,
                          const float* __restrict__ edge,
                          const float* __restrict__ edge_sp, int has_sp,
                          float* __restrict__ hsum, int nn, int total) {
    int idx = blockIdx.x * blockDim.x + threadIdx.x;
    if (idx >= total) return;                  // total = N*nn*DIM
    int E = nn * (nn - 1);
    int n = idx / (nn * DIMK);
    int v = (idx / DIMK) % nn;
    int k = idx % DIMK;
    float s = h[idx];
    for (int r = 0; r < nn - 1; ++r) {
        size_t eidx = ((size_t)(n * E + v * (nn - 1) + r)) * DIMK + k;
        s += edge[eidx];
        if (has_sp) s += edge_sp[eidx];
    }
    hsum[idx] = s;
}

// ---------------------------------------------------------------------------
// K5: inter() attention coefficients.
// cat[b,v,c] = att[2b,v,c] (c<128) | att[2b+1,v,c-128]
// s[k,v] = relu(sum_c wv[k,c]*cat[b,v,c]); softmax_v(s + 0.5*A[k,v])
// One block per b (64 blocks), computes both w12 and w21 variants.
// ---------------------------------------------------------------------------
__global__ __launch_bounds__(64) void inter_attn(
        const float* __restrict__ att, const float* __restrict__ w12,
        const float* __restrict__ w21, const float* __restrict__ A,
        float* __restrict__ att12, float* __restrict__ att21, int nn) {
    int b = blockIdx.x;
    __shared__ float s[2][VV][VV];
    int pairs = nn * nn;
    for (int p = threadIdx.x; p < 2 * pairs; p += blockDim.x) {
        int mat = p / pairs;
        int k = (p % pairs) / nn;
        int v = p % nn;
        const float* wv = (mat ? w21 : w12) + (size_t)k * (2 * DIMK);
        const float* r0 = att + (size_t)((2 * b) * nn + v) * DIMK;
        const float* r1 = att + (size_t)((2 * b + 1) * nn + v) * DIMK;
        float acc = 0.f;
        for (int c = 0; c < DIMK; ++c) acc += wv[c] * r0[c];
        for (int c = 0; c < DIMK; ++c) acc += wv[DIMK + c] * r1[c];
        s[mat][k][v] = fmaxf(acc, 0.f);
    }
    __syncthreads();
    for (int q = threadIdx.x; q < 2 * nn; q += blockDim.x) {
        int mat = q / nn;
        int k = q % nn;
        float l[VV];
        float mx = -1e30f;
        for (int v = 0; v < nn; ++v) {
            l[v] = s[mat][k][v] + 0.5f * A[k * nn + v];
            mx = fmaxf(mx, l[v]);
        }
        float den = 0.f;
        for (int v = 0; v < nn; ++v) { l[v] = __expf(l[v] - mx); den += l[v]; }
        float* dst = (mat ? att21 : att12) + (size_t)(b * nn + k) * nn;
        for (int v = 0; v < nn; ++v) dst[v] = l[v] / den;
    }
}

// ---------------------------------------------------------------------------
// K6: cross attentions.
// attc[n,v,w]  = softmax_w( tmp1[n,v,:] . s2att[n*3+w,:] )
// attw2[n,v,w] = softmax_w( tmp2[n,v,:] . s2att[n2*3+w,:] ), n2 = t2t1 remap
// ---------------------------------------------------------------------------
__global__ void cross_attn(const float* __restrict__ tmp1,
                           const float* __restrict__ tmp2,
                           const float* __restrict__ s2att,
                           float* __restrict__ attc,
                           float* __restrict__ attw2) {
    int idx = blockIdx.x * blockDim.x + threadIdx.x;   // N*V = 896
    if (idx >= NB * VV) return;
    int n = idx / VV;
    int n2 = (n < 64) ? (2 * n + 1) : (2 * (n - 64));
    float l1[WW], l2[WW];
    const float* t1 = tmp1 + (size_t)idx * DIMK;
    const float* t2 = tmp2 + (size_t)idx * DIMK;
    for (int w = 0; w < WW; ++w) {
        const float* r1 = s2att + (size_t)(n * WW + w) * DIMK;
        const float* r2 = s2att + (size_t)(n2 * WW + w) * DIMK;
        float d1 = 0.f, d2 = 0.f;
        for (int c = 0; c < DIMK; ++c) { d1 += t1[c] * r1[c]; d2 += t2[c] * r2[c]; }
        l1[w] = d1; l2[w] = d2;
    }
    float m1 = fmaxf(fmaxf(l1[0], l1[1]), l1[2]);
    float m2 = fmaxf(fmaxf(l2[0], l2[1]), l2[2]);
    float s1 = 0.f, s2 = 0.f;
    for (int w = 0; w < WW; ++w) { l1[w] = __expf(l1[w] - m1); s1 += l1[w]; }
    for (int w = 0; w < WW; ++w) { l2[w] = __expf(l2[w] - m2); s2 += l2[w]; }
    for (int w = 0; w < WW; ++w) {
        attc [(size_t)idx * WW + w] = l1[w] / s1;
        attw2[(size_t)idx * WW + w] = l2[w] / s2;
    }
}

// ---------------------------------------------------------------------------
// K7: fused s1_t1t2 stream (output #2).
// out[n,dd2,tt2,v] = concat([x1a,x2a],0)[n,v, dd2*T+tt2]; with p = dd2*T+tt2,
// x-space coords tt = p>>7, dd = p&127 (T=256, D=128).
// x1a = x_even + att21 @ x_odd ; x2a = x_odd + att12 @ x_even.
// Thread per (b,p), writes halves n=b and n=b+64 (they share x reads).
// ---------------------------------------------------------------------------
__global__ __launch_bounds__(256) void stream_s1(
        const float* __restrict__ x, const float* __restrict__ a12,
        const float* __restrict__ a21, float* __restrict__ out) {
    int b = blockIdx.y;
    __shared__ float s12[VV * VV], s21[VV * VV];
    if (threadIdx.x < VV * VV) {
        s12[threadIdx.x] = a12[b * VV * VV + threadIdx.x];
        s21[threadIdx.x] = a21[b * VV * VV + threadIdx.x];
    }
    __syncthreads();
    int p = blockIdx.x * 256 + threadIdx.x;           // 0..32767
    int tt = p >> 7, dd = p & 127;                    // input coords
    const float* pe = x + ((size_t)((2 * b) * DD + dd) * TT + tt) * VV;
    const float* po = x + ((size_t)((2 * b + 1) * DD + dd) * TT + tt) * VV;
    float xe[VV], xo[VV];
    #pragma unroll
    for (int v = 0; v < VV; ++v) { xe[v] = pe[v]; xo[v] = po[v]; }
    int dd2 = p >> 8, tt2 = p & 255;                  // output coords
    float* oa = out + ((size_t)(b * DD + dd2) * TT + tt2) * VV;
    float* ob = out + ((size_t)((b + 64) * DD + dd2) * TT + tt2) * VV;
    #pragma unroll
    for (int v = 0; v < VV; ++v) {
        float sA = xe[v], sB = xo[v];
        #pragma unroll
        for (int u = 0; u < VV; ++u) {
            sA += s21[v * VV + u] * xo[u];
            sB += s12[v * VV + u] * xe[u];
        }
        oa[v] = sA;
        ob[v] = sB;
    }
}

// ---------------------------------------------------------------------------
// K8: fused glb+glb2 stream (output #1).
// For g = dd*256+tt (output coords dd=g>>8,tt=g&255), x-space tt_p=g>>7,
// dd_p=g&127.  A1 = x1a vec, A2 = x2a vec (3 floats each):
//   n=b   : out = attc[b]   @ A1 + attw2[b]   @ A2
//   n=b+64: out = attc[b+64]@ A2 + attw2[b+64]@ A1
// ---------------------------------------------------------------------------
__global__ __launch_bounds__(256) void stream_cross(
        const float* __restrict__ x, const float* __restrict__ a12,
        const float* __restrict__ a21, const float* __restrict__ attc,
        const float* __restrict__ attw2, float* __restrict__ out) {
    int b = blockIdx.y;
    __shared__ float s12[WW * WW], s21[WW * WW];
    __shared__ float cA[VV * WW], cB[VV * WW], wA[VV * WW], wB[VV * WW];
    if (threadIdx.x < WW * WW) {
        s12[threadIdx.x] = a12[b * WW * WW + threadIdx.x];
        s21[threadIdx.x] = a21[b * WW * WW + threadIdx.x];
    }
    if (threadIdx.x < VV * WW) {
        int t = threadIdx.x;
        cA[t] = attc [(size_t)b * VV * WW + t];
        cB[t] = attc [(size_t)(b + 64) * VV * WW + t];
        wA[t] = attw2[(size_t)b * VV * WW + t];
        wB[t] = attw2[(size_t)(b + 64) * VV * WW + t];
    }
    __syncthreads();
    int g = blockIdx.x * 256 + threadIdx.x;           // 0..32767
    int tt_p = g >> 7, dd_p = g & 127;                // x-space coords
    const float* pe = x + ((size_t)((2 * b) * DD + dd_p) * TT + tt_p) * WW;
    const float* po = x + ((size_t)((2 * b + 1) * DD + dd_p) * TT + tt_p) * WW;
    float xe[WW], xo[WW], A1[WW], A2[WW];
    #pragma unroll
    for (int w = 0; w < WW; ++w) { xe[w] = pe[w]; xo[w] = po[w]; }
    #pragma unroll
    for (int w = 0; w < WW; ++w) {
        float t1 = xe[w], t2 = xo[w];
        #pragma unroll
        for (int u = 0; u < WW; ++u) {
            t1 += s21[w * WW + u] * xo[u];
            t2 += s12[w * WW + u] * xe[u];
        }
        A1[w] = t1;                                   // x1a
        A2[w] = t2;                                   // x2a
    }
    int dd = g >> 8, tt = g & 255;                    // output coords
    float* oa = out + ((size_t)(b * DD + dd) * TT + tt) * VV;
    float* ob = out + ((size_t)((b + 64) * DD + dd) * TT + tt) * VV;
    #pragma unroll
    for (int v = 0; v < VV; ++v) {
        float ra = 0.f, rb = 0.f;
        #pragma unroll
        for (int w = 0; w < WW; ++w) {
            ra += cA[v * WW + w] * A1[w] + wA[v * WW + w] * A2[w];
            rb += cB[v * WW + w] * A2[w] + wB[v * WW + w] * A1[w];
        }
        oa[v] = ra;
        ob[v] = rb;
    }
}

// ---------------------------------------------------------------------------
// Host-side orchestration
// ---------------------------------------------------------------------------
extern "C" void kernel_launch(void* const* d_in, const int* in_sizes, int n_in,
                              void* d_out, int out_size, void* d_ws, size_t ws_size,
                              hipStream_t stream) {
    const float* x_s2   = (const float*)d_in[0];
    const float* x_s1   = (const float*)d_in[1];
    const float* A_s1   = (const float*)d_in[9];   // (7,7)
    const float* A_s2   = (const float*)d_in[10];  // (3,3)
    const float* weight  = (const float*)d_in[11];
    const float* weight2 = (const float*)d_in[12];
    const float* w_t12t2    = (const float*)d_in[13];
    const float* w_t22t1    = (const float*)d_in[14];
    const float* w_t12t2_s2 = (const float*)d_in[15];
    const float* w_t22t1_s2 = (const float*)d_in[16];
    const float* Win_s1   = (const float*)d_in[17];
    const float* Wedge_s1 = (const float*)d_in[18];
    const float* Wout_s1  = (const float*)d_in[19];
    const float* Wedge_sp = (const float*)d_in[20];
    const float* Win_s2   = (const float*)d_in[21];
    const float* Wedge_s2 = (const float*)d_in[22];
    const float* Wout_s2  = (const float*)d_in[23];

    const int E1 = VV * (VV - 1);   // 42
    const int E2 = WW * (WW - 1);   // 6
    const size_t R1 = NB * VV;      // 896 rows
    const size_t R2 = NB * WW;      // 384 rows
    const size_t RE1 = (size_t)NB * E1;  // 5376 rows
    const size_t RE2 = (size_t)NB * E2;  // 768 rows

    float* ws = (float*)d_ws;
    size_t off = 0;
    float* xm1   = ws + off; off += R1 * DIMK;
    float* xm2   = ws + off; off += R2 * DIMK;
    float* h1    = ws + off; off += R1 * DIMK;
    float* h2s   = ws + off; off += R2 * DIMK;
    float* ei1   = ws + off; off += RE1 * DIMK;
    float* ei2   = ws + off; off += RE2 * DIMK;
    float* e1    = ws + off; off += RE1 * DIMK;
    float* esp   = ws + off; off += RE1 * DIMK;
    float* e2    = ws + off; off += RE2 * DIMK;
    float* hs1   = ws + off; off += R1 * DIMK;
    float* hs2   = ws + off; off += R2 * DIMK;
    float* att1  = ws + off; off += R1 * DIMK;   // x_s1_att
    float* att2s = ws + off; off += R2 * DIMK;   // x_s2_att
    float* tmp1  = ws + off; off += R1 * DIMK;
    float* tmp2  = ws + off; off += R1 * DIMK;
    float* a12_1 = ws + off; off += 64 * VV * VV;
    float* a21_1 = ws + off; off += 64 * VV * VV;
    float* a12_2 = ws + off; off += 64 * WW * WW;
    float* a21_2 = ws + off; off += 64 * WW * WW;
    float* attc  = ws + off; off += (size_t)NB * VV * WW;
    float* attw2 = ws + off; off += (size_t)NB * VV * WW;
    (void)ws_size; (void)in_sizes; (void)n_in; (void)out_size;

    float* out1 = (float*)d_out;                                   // glb+glb2
    float* out2 = (float*)d_out + (size_t)NB * DD * TT * VV;       // s1_t1t2

    // 1) means
    {
        int tot1 = NB * DD * VV, tot2 = NB * DD * WW;
        mean_over_t<<<(tot1 + 255) / 256, 256, 0, stream>>>(x_s1, xm1, VV, tot1);
        mean_over_t<<<(tot2 + 255) / 256, 256, 0, stream>>>(x_s2, xm2, WW, tot2);
    }
    // 2) h = relu(xm @ Win)
    gemm_wmma_f32<<<dim3(R1 / 16, DIMK / 16), 32, 0, stream>>>(xm1, Win_s1, h1, (int)R1, DD, DIMK, 1);
    gemm_wmma_f32<<<dim3(R2 / 16, DIMK / 16), 32, 0, stream>>>(xm2, Win_s2, h2s, (int)R2, DD, DIMK, 1);
    // 3) edge inputs (recv*send); relrec_speed == relrec_s1 so ei1 is shared
    {
        int tot1 = (int)(RE1 * DIMK), tot2 = (int)(RE2 * DIMK);
        edge_build<<<(tot1 + 255) / 256, 256, 0, stream>>>(h1, ei1, VV, tot1);
        edge_build<<<(tot2 + 255) / 256, 256, 0, stream>>>(h2s, ei2, WW, tot2);
    }
    // 4) edge = relu(ei @ Wedge)
    gemm_wmma_f32<<<dim3(RE1 / 16, DIMK / 16), 32, 0, stream>>>(ei1, Wedge_s1, e1,  (int)RE1, DIMK, DIMK, 1);
    gemm_wmma_f32<<<dim3(RE1 / 16, DIMK / 16), 32, 0, stream>>>(ei1, Wedge_sp, esp, (int)RE1, DIMK, DIMK, 1);
    gemm_wmma_f32<<<dim3(RE2 / 16, DIMK / 16), 32, 0, stream>>>(ei2, Wedge_s2, e2,  (int)RE2, DIMK, DIMK, 1);
    // 5) hsum = h + receiver-aggregated edges
    {
        int tot1 = (int)(R1 * DIMK), tot2 = (int)(R2 * DIMK);
        agg_edges<<<(tot1 + 255) / 256, 256, 0, stream>>>(h1, e1, esp, 1, hs1, VV, tot1);
        agg_edges<<<(tot2 + 255) / 256, 256, 0, stream>>>(h2s, e2, e2, 0, hs2, WW, tot2);
    }
    // 6) x_att = relu(hsum @ Wout)
    gemm_wmma_f32<<<dim3(R1 / 16, DIMK / 16), 32, 0, stream>>>(hs1, Wout_s1, att1,  (int)R1, DIMK, DIMK, 1);
    gemm_wmma_f32<<<dim3(R2 / 16, DIMK / 16), 32, 0, stream>>>(hs2, Wout_s2, att2s, (int)R2, DIMK, DIMK, 1);
    // 7) inter() attention coefficients
    inter_attn<<<64, 64, 0, stream>>>(att1,  w_t12t2,    w_t22t1,    A_s1, a12_1, a21_1, VV);
    inter_attn<<<64, 64, 0, stream>>>(att2s, w_t12t2_s2, w_t22t1_s2, A_s2, a12_2, a21_2, WW);
    // 8) tmp = x_s1_att @ weight{,2}  (no relu)
    gemm_wmma_f32<<<dim3(R1 / 16, DIMK / 16), 32, 0, stream>>>(att1, weight,  tmp1, (int)R1, DIMK, DIMK, 0);
    gemm_wmma_f32<<<dim3(R1 / 16, DIMK / 16), 32, 0, stream>>>(att1, weight2, tmp2, (int)R1, DIMK, DIMK, 0);
    // 9) cross softmax attentions
    cross_attn<<<(NB * VV + 255) / 256, 256, 0, stream>>>(tmp1, tmp2, att2s, attc, attw2);
    // 10) fused big streams
    stream_s1<<<dim3(128, 64), 256, 0, stream>>>(x_s1, a12_1, a21_1, out2);
    stream_cross<<<dim3(128, 64), 256, 0, stream>>>(x_s2, a12_2, a21_2, attc, attw2, out1);
}